// Grid_84885733638339
// MI455X (gfx1250) — compile-verified
//
#include <hip/hip_runtime.h>
#include <hip/hip_bf16.h>
#include <math.h>

#define ALPHA   0.1f
#define TWO_PI  6.2831853071795864f
#define PI_F    3.14159274f
#define PIO2_F  1.57079637f
#define INV_PI  0.3183098861837907f
#define ENV_H   512
#define ENV_W   1024
#define BLK     256

// Deterministic per-(point,sample) uniform in [0,1): PCG output hash.
__device__ __forceinline__ float rnd01(unsigned v) {
    v = v * 747796405u + 2891336453u;
    unsigned w = ((v >> ((v >> 28u) + 4u)) ^ v) * 277803737u;
    w = (w >> 22u) ^ w;
    return (float)(w >> 8) * (1.0f / 16777216.0f);
}

// Raw hardware transcendentals (~1 ulp). All inputs in this kernel are
// clamped non-negative and well inside the normal range, so the IEEE
// denormal-fixup expansions (8-10 VALU ops each) are pure waste.
__device__ __forceinline__ float fast_rcp (float x) { return __builtin_amdgcn_rcpf(x);  }
__device__ __forceinline__ float fast_sqrt(float x) { return __builtin_amdgcn_sqrtf(x); }
__device__ __forceinline__ float fast_rsq (float x) { return __builtin_amdgcn_rsqf(x);  }

// Quadrant-folded minimax atan2, ~1e-5 rad absolute error.
__device__ __forceinline__ float fast_atan2f(float y, float x) {
    float ax = fabsf(x), ay = fabsf(y);
    float mx = fmaxf(ax, ay);
    float mn = fminf(ax, ay);
    float a  = (mx > 0.0f) ? mn * fast_rcp(mx) : 0.0f;   // in [0,1]
    float s  = a * a;
    float p  = fmaf(s, fmaf(s, fmaf(s, fmaf(s, 0.0208351f, -0.0851330f),
                                    0.1801410f), -0.3302995f), 0.9998660f);
    float t  = a * p;
    t = (ay > ax) ? (PIO2_F - t) : t;
    t = (x < 0.0f) ? (PI_F - t) : t;
    return copysignf(t, y);
}

// Abramowitz-Stegun style acos, ~7e-5 rad error; input pre-clamped to (-1,1).
__device__ __forceinline__ float fast_acosf(float x) {
    float ax = fabsf(x);
    float p = fmaf(ax, fmaf(ax, fmaf(ax, -0.0187293f, 0.0742610f),
                            -0.2121144f), 1.5707288f);
    float r = p * fast_sqrt(1.0f - ax);
    return (x < 0.0f) ? (PI_F - r) : r;
}

__global__ __launch_bounds__(BLK)
void ggx_env_kernel(const float* __restrict__ dir,
                    const int*   __restrict__ spp_ptr,
                    const float* __restrict__ env,
                    float*       __restrict__ out,
                    int n)
{
    const int tid = threadIdx.x;
    const int i   = blockIdx.x * BLK + tid;
    if (i >= n) return;

    float p0, p1, p2, p3;
#if defined(__gfx1250__) && __has_builtin(__builtin_amdgcn_global_load_async_to_lds_b128) && __has_builtin(__builtin_amdgcn_s_wait_asynccnt)
    // CDNA5 async vmem->LDS path (tracked by ASYNCcnt): stage this block's dir tile.
    {
        __shared__ __align__(16) float s_dir[BLK * 4];
        typedef int v4i __attribute__((vector_size(16)));
        typedef __attribute__((address_space(1))) v4i* gv4_t;
        typedef __attribute__((address_space(3))) v4i* lv4_t;
        __builtin_amdgcn_global_load_async_to_lds_b128(
            (gv4_t)(dir + (size_t)i * 4), (lv4_t)(&s_dir[tid * 4]), 0, 0);
        __builtin_amdgcn_s_wait_asynccnt(0);
        p0 = s_dir[tid * 4 + 0];
        p1 = s_dir[tid * 4 + 1];
        p2 = s_dir[tid * 4 + 2];
        p3 = s_dir[tid * 4 + 3];
    }
#else
    {
        float4 p = ((const float4*)dir)[i];
        p0 = p.x; p1 = p.y; p2 = p.z; p3 = p.w;
    }
#endif

    const int spp = *spp_ptr;

    // ---- per-point precompute (sample-invariant) ----
    // normal from canonical square
    float nz = 2.0f * p0 - 1.0f;
    float stn = fast_sqrt(fmaxf(1.0f - nz * nz, 0.0f));
    float snp, cnp; __sincosf(TWO_PI * p1, &snp, &cnp);
    float nx = stn * cnp, ny = stn * snp;

    // wi from hemisphere canonical
    float wiz = 1.0f - p2;
    float sti = fast_sqrt(fmaxf(1.0f - wiz * wiz, 0.0f));
    float sip, cip; __sincosf(TWO_PI * p3, &sip, &cip);
    float wix = sti * cip, wiy = sti * sip;

    // rotate_align(up=(0,0,1), nrm): d_world = wo^T * M
    float tchk = nz + 0.9999f;
    float check = (tchk > 0.0f) ? 1.0f : ((tchk < 0.0f) ? 0.0f : 0.5f);
    float kk = fast_rcp(1.0f + nz * check);
    float r00 = ny * ny * kk + nz;
    float r01 = -nx * ny * kk;           // == r10
    float r11 = nx * nx * kk + nz;

    // GGX VNDF frame: depends only on wi and alpha -> hoist out of spp loop
    float vhx = wix * ALPHA, vhy = wiy * ALPHA, vhz = wiz;
    float invn = fast_rsq(vhx * vhx + vhy * vhy + vhz * vhz);
    vhx *= invn; vhy *= invn; vhz *= invn;
    float lensq = vhx * vhx + vhy * vhy;
    float il = fast_rsq(fmaxf(lensq, 1e-12f));
    float T1x, T1y, T1z;
    if (lensq > 1e-7f) { T1x = -vhy * il; T1y = vhx * il; T1z = 0.0f; }
    else               { T1x = 1.0f;      T1y = 0.0f;     T1z = 0.0f; }
    float T2x = vhy * T1z - vhz * T1y;
    float T2y = vhz * T1x - vhx * T1z;
    float T2z = vhx * T1y - vhy * T1x;
    float svh = 0.5f * (1.0f + vhz);

    const float ETA2[3] = {1.3456f * 1.3456f, 0.96521f * 0.96521f, 0.61722f * 0.61722f};
    const float KAP2[3] = {7.4746f * 7.4746f, 6.3995f * 6.3995f, 5.3031f * 5.3031f};

    float accx = 0.0f, accy = 0.0f, accz = 0.0f;

#pragma unroll 2
    for (int it = 0; it < spp; ++it) {
        unsigned seed = (unsigned)i * 0x9E3779B9u ^ ((unsigned)it * 0x632BE59Bu);
        float u1 = rnd01(seed * 2u + 0x68BC21EBu);
        float u2 = rnd01(seed * 2u + 0x02E5BE93u);

        // sample_ggx_vndf
        float r = fast_sqrt(u1);
        float sph, cph; __sincosf(TWO_PI * u2, &sph, &cph);
        float t1 = r * cph;
        float t2 = r * sph;
        t2 = (1.0f - svh) * fast_sqrt(fmaxf(1.0f - t1 * t1, 0.0f)) + svh * t2;
        float t3 = fast_sqrt(fmaxf(1.0f - t1 * t1 - t2 * t2, 0.0f));
        float nhx = t1 * T1x + t2 * T2x + t3 * vhx;
        float nhy = t1 * T1y + t2 * T2y + t3 * vhy;
        float nhz = t1 * T1z + t2 * T2z + t3 * vhz;
        float mx = ALPHA * nhx, my = ALPHA * nhy, mz = fmaxf(nhz, 1e-6f);
        float imn = fast_rsq(mx * mx + my * my + mz * mz);
        mx *= imn; my *= imn; mz *= imn;

        float dwm = wix * mx + wiy * my + wiz * mz;
        float wox = 2.0f * dwm * mx - wix;
        float woy = 2.0f * dwm * my - wiy;
        float woz = 2.0f * dwm * mz - wiz;
        bool valid = (woz > 0.0f) && (dwm > 0.0f);

        // smith_g1(wo, m, alpha)
        float tan2 = (wox * wox + woy * woy) * fast_rcp(fmaxf(woz * woz, 1e-12f));
        float g = 2.0f * fast_rcp(1.0f + fast_sqrt(1.0f + ALPHA * ALPHA * tan2));
        float wodm = wox * mx + woy * my + woz * mz;
        g = ((wodm * woz) > 0.0f && valid) ? g : 0.0f;

        // fresnel_conductor(dwm), 3 channels
        float ct = fminf(fmaxf(dwm, 0.0f), 1.0f);
        float c2 = ct * ct, s2 = 1.0f - c2;
        float F[3];
#pragma unroll
        for (int c = 0; c < 3; ++c) {
            float t0   = ETA2[c] - KAP2[c] - s2;
            float a2b2 = fast_sqrt(fmaxf(t0 * t0 + 4.0f * ETA2[c] * KAP2[c], 1e-12f));
            float tt1  = a2b2 + c2;
            float a    = fast_sqrt(fmaxf(0.5f * (a2b2 + t0), 1e-12f));
            float tt2  = 2.0f * a * ct;
            float rs   = (tt1 - tt2) * fast_rcp(tt1 + tt2);
            float tt3  = c2 * a2b2 + s2 * s2;
            float tt4  = tt2 * s2;
            float rp   = rs * (tt3 - tt4) * fast_rcp(tt3 + tt4);
            F[c] = 0.5f * (rs + rp);
        }

        // d_world = wo^T * M
        float dxw = wox * r00 + woy * r01 - woz * nx;
        float dyw = wox * r01 + woy * r11 - woz * ny;
        float dzw = wox * nx  + woy * ny  + woz * nz;

        // envmap bilinear (wrap x, clamp y)
        float u = 0.5f * (1.0f + fast_atan2f(dxw, -dzw) * INV_PI);
        float v = fast_acosf(fminf(fmaxf(dyw, -1.0f + 1e-6f), 1.0f - 1e-6f)) * INV_PI;
        float xf = u * (float)ENV_W - 0.5f;
        float yf = v * (float)ENV_H - 0.5f;
        float x0 = floorf(xf), y0 = floorf(yf);
        float fx = xf - x0,   fy = yf - y0;
        int x0i = (int)x0 % ENV_W; if (x0i < 0) x0i += ENV_W;
        int x1i = x0i + 1;         if (x1i >= ENV_W) x1i = 0;
        int y0i = min(max((int)y0, 0), ENV_H - 1);
        int y1i = min(y0i + 1, ENV_H - 1);

        const float* e00 = env + ((size_t)y0i * ENV_W + x0i) * 3;
        const float* e01 = env + ((size_t)y0i * ENV_W + x1i) * 3;
        const float* e10 = env + ((size_t)y1i * ENV_W + x0i) * 3;
        const float* e11 = env + ((size_t)y1i * ENV_W + x1i) * 3;
        float w00 = (1.0f - fx) * (1.0f - fy);
        float w01 = fx * (1.0f - fy);
        float w10 = (1.0f - fx) * fy;
        float w11 = fx * fy;

        float er = w00 * e00[0] + w01 * e01[0] + w10 * e10[0] + w11 * e11[0];
        float eg = w00 * e00[1] + w01 * e01[1] + w10 * e10[1] + w11 * e11[1];
        float eb = w00 * e00[2] + w01 * e01[2] + w10 * e10[2] + w11 * e11[2];

        accx = fmaf(F[0] * g, er, accx);
        accy = fmaf(F[1] * g, eg, accy);
        accz = fmaf(F[2] * g, eb, accz);
    }

    float inv_spp = 1.0f / (float)spp;
    out[(size_t)i * 3 + 0] = accx * inv_spp;
    out[(size_t)i * 3 + 1] = accy * inv_spp;
    out[(size_t)i * 3 + 2] = accz * inv_spp;
}

extern "C" void kernel_launch(void* const* d_in, const int* in_sizes, int n_in,
                              void* d_out, int out_size, void* d_ws, size_t ws_size,
                              hipStream_t stream) {
    const float* dir = (const float*)d_in[0];   // (n,4) float32
    const int*   spp = (const int*)d_in[1];     // scalar int on device
    const float* env = (const float*)d_in[2];   // (512,1024,3) float32
    float*       out = (float*)d_out;           // (n,3) float32

    int n = in_sizes[0] / 4;
    dim3 block(BLK);
    dim3 grid((n + BLK - 1) / BLK);
    ggx_env_kernel<<<grid, block, 0, stream>>>(dir, spp, env, out, n);
}